// Model_29197187678419
// MI455X (gfx1250) — compile-verified
//
#include <hip/hip_runtime.h>
#include <hip/hip_bf16.h>

typedef float v2f __attribute__((ext_vector_type(2)));
typedef float v8f __attribute__((ext_vector_type(8)));

#define NEG_SLOPE 0.01f
#define NGRAPHS 256

__device__ __forceinline__ float lrelu(float v) { return v > 0.f ? v : NEG_SLOPE * v; }

// D = A(16x4 f32) * B(4x16 f32) + C(16x16 f32), wave32 WMMA
__device__ __forceinline__ v8f wmma4(v2f a, v2f b, v8f c) {
  return __builtin_amdgcn_wmma_f32_16x16x4_f32(false, a, false, b, (short)0, c, false, false);
}

// ---------------- layer 1: scalar-feature edge scatter + degree ----------------
__global__ void edge_agg1(const int* __restrict__ src, const int* __restrict__ dst,
                          const float* __restrict__ x, float* __restrict__ agg1,
                          float* __restrict__ cnt, int E) {
  int e = blockIdx.x * blockDim.x + threadIdx.x;
  if (e >= E) return;
  int s = src[e], d = dst[e];
  atomicAdd(&agg1[d], x[s]);
  atomicAdd(&cnt[d], 1.0f);
}

// h1[n][j] = leaky(agg1[n]*W1l[j] + b1[j] + x[n]*W1r[j])
__global__ void node1(const float* __restrict__ x, const float* __restrict__ agg1,
                      const float* __restrict__ W1l, const float* __restrict__ b1,
                      const float* __restrict__ W1r, float* __restrict__ h1, int N) {
  int idx = blockIdx.x * blockDim.x + threadIdx.x;
  if (idx >= N * 32) return;
  int n = idx >> 5, j = idx & 31;
  h1[idx] = lrelu(agg1[n] * W1l[j] + b1[j] + x[n] * W1r[j]);
}

// ---------------- 32-wide edge scatter: one wave per edge (coalesced) ----------------
__global__ void edge_agg32(const int* __restrict__ src, const int* __restrict__ dst,
                           const float* __restrict__ h, float* __restrict__ agg, int E) {
  int idx = blockIdx.x * blockDim.x + threadIdx.x;
  if (idx >= E * 32) return;
  int e = idx >> 5, f = idx & 31;   // 32 lanes of a wave share e -> sgpr broadcast
  atomicAdd(&agg[(size_t)dst[e] * 32 + f], h[(size_t)src[e] * 32 + f]);
}

// ---------------- WMMA node GEMM: hout = leaky([agg|hin] @ [Wl;Wr] + b), H=32 ----------------
__global__ void sage_node_wmma(const float* __restrict__ agg, const float* __restrict__ hin,
                               const float* __restrict__ Wl, const float* __restrict__ Wr,
                               const float* __restrict__ b, float* __restrict__ hout,
                               int nTiles) {
  int wave = (blockIdx.x * blockDim.x + threadIdx.x) >> 5;
  int lane = threadIdx.x & 31;
  if (wave >= nTiles) return;
  int m = lane & 15, kh = lane >> 4;       // A: row m; B/C: col m; kh = K-half / M-half
  size_t row = (size_t)wave * 16 + m;

  v8f c0 = {}, c1 = {};
  const float* Ap[2] = { agg + row * 32, hin + row * 32 };
  const float* Wp[2] = { Wl, Wr };
  for (int p = 0; p < 2; ++p) {
    const float* A = Ap[p];
    const float* W = Wp[p];
#pragma unroll
    for (int kk = 0; kk < 8; ++kk) {       // K = 32 in steps of 4
      int k0 = kk * 4 + kh * 2;
      v2f a = *(const v2f*)(A + k0);       // A[m][k0], A[m][k0+1]
      v2f b0, b1;
      b0.x = W[k0 * 32 + m];       b0.y = W[(k0 + 1) * 32 + m];
      b1.x = W[k0 * 32 + 16 + m];  b1.y = W[(k0 + 1) * 32 + 16 + m];
      c0 = wmma4(a, b0, c0);
      c1 = wmma4(a, b1, c1);
    }
  }
  float bias0 = b[m], bias1 = b[16 + m];
#pragma unroll
  for (int r = 0; r < 8; ++r) {
    size_t orow = (size_t)wave * 16 + r + 8 * kh;
    hout[orow * 32 + m]      = lrelu(c0[r] + bias0);
    hout[orow * 32 + 16 + m] = lrelu(c1[r] + bias1);
  }
}

// ---------------- fused layer3 (mean) + layer4 upscale, all WMMA ----------------
// h3 = leaky([agg/cnt | hin] @ [W3l;W3r] + b3)   (16x32 tile, via LDS)
// h4 = leaky(h3 @ Wu + bu)                        (16x128 tile)
__global__ void sage_node3_up_wmma(const float* __restrict__ agg, const float* __restrict__ cnt,
                                   const float* __restrict__ hin,
                                   const float* __restrict__ W3l, const float* __restrict__ W3r,
                                   const float* __restrict__ b3,
                                   const float* __restrict__ Wu, const float* __restrict__ bu,
                                   float* __restrict__ h4, int nTiles) {
  __shared__ float lds[8][16][33];          // per-wave 16x32 tile, padded
  int wslot = threadIdx.x >> 5;
  int wave = (blockIdx.x * blockDim.x + threadIdx.x) >> 5;
  int lane = threadIdx.x & 31;
  if (wave >= nTiles) return;
  int m = lane & 15, kh = lane >> 4;
  size_t row = (size_t)wave * 16 + m;
  float inv = 1.0f / fmaxf(cnt[row], 1.0f); // mean aggregation

  v8f c0 = {}, c1 = {};
  for (int p = 0; p < 2; ++p) {
    const float* A = (p == 0 ? agg : hin) + row * 32;
    const float* W = (p == 0 ? W3l : W3r);
    float sc = (p == 0 ? inv : 1.0f);
#pragma unroll
    for (int kk = 0; kk < 8; ++kk) {
      int k0 = kk * 4 + kh * 2;
      v2f a = *(const v2f*)(A + k0);
      a.x *= sc; a.y *= sc;
      v2f b0, b1;
      b0.x = W[k0 * 32 + m];       b0.y = W[(k0 + 1) * 32 + m];
      b1.x = W[k0 * 32 + 16 + m];  b1.y = W[(k0 + 1) * 32 + 16 + m];
      c0 = wmma4(a, b0, c0);
      c1 = wmma4(a, b1, c1);
    }
  }
  // h3 tile -> LDS (C layout -> row-major), bias + leaky
  float bias0 = b3[m], bias1 = b3[16 + m];
#pragma unroll
  for (int r = 0; r < 8; ++r) {
    lds[wslot][r + 8 * kh][m]      = lrelu(c0[r] + bias0);
    lds[wslot][r + 8 * kh][16 + m] = lrelu(c1[r] + bias1);
  }
  // per-wave LDS: DS ops are in-order within a wave; compiler inserts s_wait_dscnt

  // second GEMM: [16x32] @ Wu[32x128]
  v8f acc[8] = {};
#pragma unroll
  for (int kk = 0; kk < 8; ++kk) {
    int k0 = kk * 4 + kh * 2;
    v2f a; a.x = lds[wslot][m][k0]; a.y = lds[wslot][m][k0 + 1];
#pragma unroll
    for (int t = 0; t < 8; ++t) {
      int col = t * 16 + m;
      v2f bb; bb.x = Wu[k0 * 128 + col]; bb.y = Wu[(k0 + 1) * 128 + col];
      acc[t] = wmma4(a, bb, acc[t]);
    }
  }
#pragma unroll
  for (int t = 0; t < 8; ++t) {
    int col = t * 16 + m;
    float bias = bu[col];
#pragma unroll
    for (int r = 0; r < 8; ++r) {
      size_t orow = (size_t)wave * 16 + r + 8 * kh;
      h4[orow * 128 + col] = lrelu(acc[t][r] + bias);
    }
  }
}

// ---------------- pooling ----------------
__global__ void fill_neg(float* __restrict__ p, int n) {
  int i = blockIdx.x * blockDim.x + threadIdx.x;
  if (i < n) p[i] = -3.402823466e+38f;
}

__device__ __forceinline__ void atomicMaxFloat(float* addr, float val) {
  // ordered-int trick: positives via signed max, negatives via unsigned min
  if (val >= 0.f) atomicMax((int*)addr, __float_as_int(val));
  else            atomicMin((unsigned int*)addr, __float_as_uint(val));
}

__global__ void pool_kernel(const float* __restrict__ h4, const int* __restrict__ batch,
                            float* __restrict__ addp, float* __restrict__ maxp,
                            float* __restrict__ gcnt, int N) {
  int idx = blockIdx.x * blockDim.x + threadIdx.x;
  if (idx >= N * 128) return;
  int n = idx >> 7, f = idx & 127;
  int g = batch[n];
  float v = h4[idx];
  atomicAdd(&addp[g * 128 + f], v);
  atomicMaxFloat(&maxp[g * 128 + f], v);
  if (f == 0) atomicAdd(&gcnt[g], 1.0f);
}

__global__ void build_z(const float* __restrict__ addp, const float* __restrict__ maxp,
                        const float* __restrict__ gcnt, float* __restrict__ z) {
  int idx = blockIdx.x * blockDim.x + threadIdx.x;
  if (idx >= NGRAPHS * 384) return;
  int g = idx / 384, k = idx % 384;
  float v;
  if (k < 128)      v = addp[g * 128 + k] / fmaxf(gcnt[g], 1.0f);
  else if (k < 256) v = maxp[g * 128 + (k - 128)];
  else              v = addp[g * 128 + (k - 256)];
  z[idx] = v;
}

// ---------------- final MLP head: z[256,384] @ Wf1 -> leaky -> @ Wf2[128,51] ----------------
__global__ void mlp_head_wmma(const float* __restrict__ z,
                              const float* __restrict__ Wf1, const float* __restrict__ bf1,
                              const float* __restrict__ Wf2, const float* __restrict__ bf2,
                              float* __restrict__ out) {
  __shared__ float y[16][129];              // 16 graphs x 128 features
  int tile = blockIdx.x;                    // 16 blocks x 1 wave
  int lane = threadIdx.x & 31;
  int m = lane & 15, kh = lane >> 4;
  size_t g = (size_t)tile * 16 + m;

  // GEMM1: K=384, N=128
  v8f acc[8] = {};
#pragma unroll 4
  for (int kk = 0; kk < 96; ++kk) {
    int k0 = kk * 4 + kh * 2;
    v2f a = *(const v2f*)(z + g * 384 + k0);
#pragma unroll
    for (int t = 0; t < 8; ++t) {
      int col = t * 16 + m;
      v2f bb; bb.x = Wf1[k0 * 128 + col]; bb.y = Wf1[(k0 + 1) * 128 + col];
      acc[t] = wmma4(a, bb, acc[t]);
    }
  }
#pragma unroll
  for (int t = 0; t < 8; ++t) {
    int col = t * 16 + m;
    float bias = bf1[col];
#pragma unroll
    for (int r = 0; r < 8; ++r)
      y[r + 8 * kh][col] = lrelu(acc[t][r] + bias);
  }

  // GEMM2: K=128, N=51 (padded to 64, masked)
  v8f acc2[4] = {};
#pragma unroll
  for (int kk = 0; kk < 32; ++kk) {
    int k0 = kk * 4 + kh * 2;
    v2f a; a.x = y[m][k0]; a.y = y[m][k0 + 1];
#pragma unroll
    for (int t = 0; t < 4; ++t) {
      int col = t * 16 + m;
      v2f bb;
      bb.x = (col < 51) ? Wf2[k0 * 51 + col] : 0.f;
      bb.y = (col < 51) ? Wf2[(k0 + 1) * 51 + col] : 0.f;
      acc2[t] = wmma4(a, bb, acc2[t]);
    }
  }
#pragma unroll
  for (int t = 0; t < 4; ++t) {
    int col = t * 16 + m;
    if (col < 51) {
      float bias = bf2[col];
#pragma unroll
      for (int r = 0; r < 8; ++r) {
        size_t og = (size_t)tile * 16 + r + 8 * kh;
        out[og * 51 + col] = acc2[t][r] + bias;
      }
    }
  }
}

extern "C" void kernel_launch(void* const* d_in, const int* in_sizes, int n_in,
                              void* d_out, int out_size, void* d_ws, size_t ws_size,
                              hipStream_t stream) {
  const float* x   = (const float*)d_in[0];
  const int*   ei  = (const int*)d_in[1];
  const int* batch = (const int*)d_in[2];
  const float* W1l = (const float*)d_in[3];
  const float* b1  = (const float*)d_in[4];
  const float* W1r = (const float*)d_in[5];
  const float* W2l = (const float*)d_in[6];
  const float* b2  = (const float*)d_in[7];
  const float* W2r = (const float*)d_in[8];
  const float* W3l = (const float*)d_in[9];
  const float* b3  = (const float*)d_in[10];
  const float* W3r = (const float*)d_in[11];
  const float* Wu  = (const float*)d_in[12];
  const float* bu  = (const float*)d_in[13];
  const float* Wf1 = (const float*)d_in[14];
  const float* bf1 = (const float*)d_in[15];
  const float* Wf2 = (const float*)d_in[16];
  const float* bf2 = (const float*)d_in[17];
  float* out = (float*)d_out;

  const int N = in_sizes[0];        // 100000 (multiple of 16)
  const int E = in_sizes[1] / 2;    // 1600000
  const int* src = ei;
  const int* dst = ei + E;
  const int nTiles = N / 16;

  // workspace layout (floats)
  float* w = (float*)d_ws;
  float* agg1 = w;  w += N;
  float* cnt  = w;  w += N;
  float* h1   = w;  w += (size_t)N * 32;
  float* agg  = w;  w += (size_t)N * 32;   // reused for layer2 & layer3 aggregation
  float* h2   = w;  w += (size_t)N * 32;
  float* h4   = w;  w += (size_t)N * 128;
  float* addp = w;  w += NGRAPHS * 128;
  float* maxp = w;  w += NGRAPHS * 128;
  float* gcnt = w;  w += NGRAPHS;
  float* z    = w;  w += NGRAPHS * 384;

  const int TB = 256;

  // ---- layer 1 ----
  hipMemsetAsync(agg1, 0, (size_t)2 * N * sizeof(float), stream);   // agg1 + cnt
  edge_agg1<<<(E + TB - 1) / TB, TB, 0, stream>>>(src, dst, x, agg1, cnt, E);
  node1<<<((size_t)N * 32 + TB - 1) / TB, TB, 0, stream>>>(x, agg1, W1l, b1, W1r, h1, N);

  // ---- layer 2 ----
  hipMemsetAsync(agg, 0, (size_t)N * 32 * sizeof(float), stream);
  edge_agg32<<<((size_t)E * 32 + TB - 1) / TB, TB, 0, stream>>>(src, dst, h1, agg, E);
  sage_node_wmma<<<(nTiles + 7) / 8, TB, 0, stream>>>(agg, h1, W2l, W2r, b2, h2, nTiles);

  // ---- layer 3 (mean) + upscale ----
  hipMemsetAsync(agg, 0, (size_t)N * 32 * sizeof(float), stream);
  edge_agg32<<<((size_t)E * 32 + TB - 1) / TB, TB, 0, stream>>>(src, dst, h2, agg, E);
  sage_node3_up_wmma<<<(nTiles + 7) / 8, TB, 0, stream>>>(agg, cnt, h2, W3l, W3r, b3,
                                                          Wu, bu, h4, nTiles);

  // ---- pooling ----
  hipMemsetAsync(addp, 0, (size_t)NGRAPHS * 128 * sizeof(float), stream);
  hipMemsetAsync(gcnt, 0, (size_t)NGRAPHS * sizeof(float), stream);
  fill_neg<<<(NGRAPHS * 128 + TB - 1) / TB, TB, 0, stream>>>(maxp, NGRAPHS * 128);
  pool_kernel<<<((size_t)N * 128 + TB - 1) / TB, TB, 0, stream>>>(h4, batch, addp, maxp, gcnt, N);
  build_z<<<(NGRAPHS * 384 + TB - 1) / TB, TB, 0, stream>>>(addp, maxp, gcnt, z);

  // ---- MLP head ----
  mlp_head_wmma<<<NGRAPHS / 16, 32, 0, stream>>>(z, Wf1, bf1, Wf2, bf2, out);
}